// CrossAttentionSpatialWeight_83880711291224
// MI455X (gfx1250) — compile-verified
//
#include <hip/hip_runtime.h>
#include <math.h>

#define N_NODES   4096
#define GENE_D    512
#define HID       256
#define OUT_D     256

typedef __attribute__((ext_vector_type(2))) float v2f;
typedef __attribute__((ext_vector_type(8))) float v8f;

#define WMMA_F32(A, B, C) \
    __builtin_amdgcn_wmma_f32_16x16x4_f32(false, (A), false, (B), (short)0, (C), false, false)

// ---------------------------------------------------------------------------
// Generic WMMA f32 GEMM:  Y[n,m] = act(X[n,kd] @ W[kd,m] + bias[m])
// One wave per 32x32 output tile = 2x2 grid of 16x16 WMMA subtiles.
// Per k-step: 4 fragment loads feed 4 wmma (1.0 load/wmma vs 2.0 untiled).
// ACT: 0 = none, 1 = tanh, 2 = relu
// ---------------------------------------------------------------------------
template <int ACT>
__global__ __launch_bounds__(32)
void gemm32_wmma(const float* __restrict__ X, const float* __restrict__ W,
                 const float* __restrict__ bias, float* __restrict__ Y,
                 int kd, int m) {
    const int lane = threadIdx.x;
    const int hgrp = lane >> 4;          // 0 or 1 (half-wave)
    const int ln   = lane & 15;
    const int colBase = blockIdx.x * 32;
    const int rowBase = blockIdx.y * 32;

    v8f acc00 = {}, acc01 = {}, acc10 = {}, acc11 = {};
    const float* xr0 = X + (size_t)(rowBase + ln) * kd;
    const float* xr1 = X + (size_t)(rowBase + 16 + ln) * kd;
    const float* wc0 = W + colBase + ln;
    const float* wc1 = W + colBase + 16 + ln;

    for (int kb = 0; kb < kd; kb += 4) {
        const int k0 = kb + 2 * hgrp;
        v2f a0 = *(const v2f*)(xr0 + k0);
        v2f a1 = *(const v2f*)(xr1 + k0);
        v2f b0, b1;
        b0.x = wc0[(size_t)k0 * m];       b0.y = wc0[(size_t)(k0 + 1) * m];
        b1.x = wc1[(size_t)k0 * m];       b1.y = wc1[(size_t)(k0 + 1) * m];
        acc00 = WMMA_F32(a0, b0, acc00);
        acc01 = WMMA_F32(a0, b1, acc01);
        acc10 = WMMA_F32(a1, b0, acc10);
        acc11 = WMMA_F32(a1, b1, acc11);
    }

    const float bv0 = bias ? bias[colBase + ln]      : 0.0f;
    const float bv1 = bias ? bias[colBase + 16 + ln] : 0.0f;

#pragma unroll
    for (int v = 0; v < 8; ++v) {
        const int i0 = rowBase + v + 8 * hgrp;        // D-row per ISA C/D layout
        const int i1 = i0 + 16;
        float r00 = acc00[v] + bv0, r01 = acc01[v] + bv1;
        float r10 = acc10[v] + bv0, r11 = acc11[v] + bv1;
        if (ACT == 1) { r00 = tanhf(r00); r01 = tanhf(r01); r10 = tanhf(r10); r11 = tanhf(r11); }
        if (ACT == 2) { r00 = fmaxf(r00, 0.0f); r01 = fmaxf(r01, 0.0f);
                        r10 = fmaxf(r10, 0.0f); r11 = fmaxf(r11, 0.0f); }
        Y[(size_t)i0 * m + colBase + ln]      = r00;
        Y[(size_t)i0 * m + colBase + 16 + ln] = r01;
        Y[(size_t)i1 * m + colBase + ln]      = r10;
        Y[(size_t)i1 * m + colBase + 16 + ln] = r11;
    }
}

// ---------------------------------------------------------------------------
// Fused logits, 32x32 tile per wave:
//   L[i][j] = (Q[i].K[j])/16 + exp(-d2/2) + cos_sim + tanh(f[i]-f[j])
// Two accumulation passes (QK^T over K=256, gene.gene^T over K=512), each
// 2x2-register-tiled. Both are A@B^T so every fragment is a contiguous
// float2 of a row-major matrix.
// ---------------------------------------------------------------------------
__global__ __launch_bounds__(32)
void logits_wmma(const float* __restrict__ Q, const float* __restrict__ K,
                 const float* __restrict__ G, const float* __restrict__ sp,
                 const float* __restrict__ sq, const float* __restrict__ inv_nrm,
                 const float* __restrict__ f, float* __restrict__ L) {
    const int lane = threadIdx.x;
    const int hgrp = lane >> 4;
    const int ln   = lane & 15;
    const int colBase = blockIdx.x * 32;
    const int rowBase = blockIdx.y * 32;

    v8f q00 = {}, q01 = {}, q10 = {}, q11 = {};
    {
        const float* qr0 = Q + (size_t)(rowBase + ln) * HID;
        const float* qr1 = Q + (size_t)(rowBase + 16 + ln) * HID;
        const float* kr0 = K + (size_t)(colBase + ln) * HID;
        const float* kr1 = K + (size_t)(colBase + 16 + ln) * HID;
        for (int kb = 0; kb < HID; kb += 4) {
            const int k0 = kb + 2 * hgrp;
            v2f a0 = *(const v2f*)(qr0 + k0);
            v2f a1 = *(const v2f*)(qr1 + k0);
            v2f b0 = *(const v2f*)(kr0 + k0);   // B[k][n] = K[col+n][k]
            v2f b1 = *(const v2f*)(kr1 + k0);
            q00 = WMMA_F32(a0, b0, q00);
            q01 = WMMA_F32(a0, b1, q01);
            q10 = WMMA_F32(a1, b0, q10);
            q11 = WMMA_F32(a1, b1, q11);
        }
    }

    v8f g00 = {}, g01 = {}, g10 = {}, g11 = {};
    {
        const float* gr0 = G + (size_t)(rowBase + ln) * GENE_D;
        const float* gr1 = G + (size_t)(rowBase + 16 + ln) * GENE_D;
        const float* gc0 = G + (size_t)(colBase + ln) * GENE_D;
        const float* gc1 = G + (size_t)(colBase + 16 + ln) * GENE_D;
        for (int kb = 0; kb < GENE_D; kb += 4) {
            const int k0 = kb + 2 * hgrp;
            v2f a0 = *(const v2f*)(gr0 + k0);
            v2f a1 = *(const v2f*)(gr1 + k0);
            v2f b0 = *(const v2f*)(gc0 + k0);
            v2f b1 = *(const v2f*)(gc1 + k0);
            g00 = WMMA_F32(a0, b0, g00);
            g01 = WMMA_F32(a0, b1, g01);
            g10 = WMMA_F32(a1, b0, g10);
            g11 = WMMA_F32(a1, b1, g11);
        }
    }

    // Column-side (j) scalars: fixed per lane, one set per column subtile.
    const int j0 = colBase + ln;
    const int j1 = j0 + 16;
    const float sqj0 = sq[j0],  sqj1 = sq[j1];
    const float fj0  = f[j0],   fj1  = f[j1];
    const float inj0 = inv_nrm[j0], inj1 = inv_nrm[j1];
    const float sj00 = sp[2 * j0], sj01 = sp[2 * j0 + 1];
    const float sj10 = sp[2 * j1], sj11 = sp[2 * j1 + 1];
    const float scale = 0.0625f;   // 1/sqrt(HID) = 1/16

#pragma unroll
    for (int v = 0; v < 8; ++v) {
#pragma unroll
        for (int di = 0; di < 2; ++di) {
            const int i = rowBase + 16 * di + v + 8 * hgrp;
            const float sqi = sq[i], fi = f[i], ini = inv_nrm[i];
            const float si0 = sp[2 * i], si1 = sp[2 * i + 1];

            const float qk0 = (di == 0) ? q00[v] : q10[v];
            const float qk1 = (di == 0) ? q01[v] : q11[v];
            const float gg0 = (di == 0) ? g00[v] : g10[v];
            const float gg1 = (di == 0) ? g01[v] : g11[v];

            const float d2a = fmaxf(sqi + sqj0 - 2.0f * (si0 * sj00 + si1 * sj01), 0.0f);
            const float d2b = fmaxf(sqi + sqj1 - 2.0f * (si0 * sj10 + si1 * sj11), 0.0f);

            L[(size_t)i * N_NODES + j0] =
                qk0 * scale + expf(-0.5f * d2a) + gg0 * ini * inj0 + tanhf(fi - fj0);
            L[(size_t)i * N_NODES + j1] =
                qk1 * scale + expf(-0.5f * d2b) + gg1 * ini * inj1 + tanhf(fi - fj1);
        }
    }
}

// ---------------------------------------------------------------------------
// Per-row stats: inv_nrm[i] = 1/max(||gene_i||,1e-12), sq[i] = ||sp_i||^2
// ---------------------------------------------------------------------------
__global__ __launch_bounds__(256)
void row_stats(const float* __restrict__ G, const float* __restrict__ sp,
               float* __restrict__ sq, float* __restrict__ inv_nrm) {
    __shared__ float red[256];
    const int row = blockIdx.x;
    const int t = threadIdx.x;
    const float* g = G + (size_t)row * GENE_D;
    const float a = g[t];
    const float b = g[t + 256];
    red[t] = a * a + b * b;
    __syncthreads();
    for (int s = 128; s > 0; s >>= 1) {
        if (t < s) red[t] += red[t + s];
        __syncthreads();
    }
    if (t == 0) {
        inv_nrm[row] = 1.0f / fmaxf(sqrtf(red[0]), 1e-12f);
        const float s0 = sp[2 * row], s1 = sp[2 * row + 1];
        sq[row] = s0 * s0 + s1 * s1;
    }
}

// si = spatial_pos @ Ws + bs   (K = 2, pure VALU)
__global__ __launch_bounds__(256)
void spatial_in(const float* __restrict__ sp, const float* __restrict__ Ws,
                const float* __restrict__ bs, float* __restrict__ si) {
    const int idx = blockIdx.x * 256 + threadIdx.x;
    if (idx >= N_NODES * HID) return;
    const int i = idx >> 8;
    const int j = idx & 255;
    si[idx] = sp[2 * i] * Ws[j] + sp[2 * i + 1] * Ws[HID + j] + bs[j];
}

// f[i] = e[i,:] . Wf + bf
__global__ __launch_bounds__(256)
void final_proj(const float* __restrict__ e, const float* __restrict__ Wf,
                const float* __restrict__ bf, float* __restrict__ f) {
    __shared__ float red[256];
    const int row = blockIdx.x;
    const int t = threadIdx.x;
    red[t] = e[(size_t)row * HID + t] * Wf[t];
    __syncthreads();
    for (int s = 128; s > 0; s >>= 1) {
        if (t < s) red[t] += red[t + s];
        __syncthreads();
    }
    if (t == 0) f[row] = red[0] + bf[0];
}

// In-place row softmax over 4096 columns (16 values/thread in registers).
__global__ __launch_bounds__(256)
void softmax_rows(float* __restrict__ L) {
    __shared__ float red[256];
    const int t = threadIdx.x;
    float* rp = L + (size_t)blockIdx.x * N_NODES;

    float vals[16];
    float mx = -INFINITY;
#pragma unroll
    for (int e = 0; e < 16; ++e) {
        vals[e] = rp[t + e * 256];
        mx = fmaxf(mx, vals[e]);
    }
    red[t] = mx;
    __syncthreads();
    for (int s = 128; s > 0; s >>= 1) {
        if (t < s) red[t] = fmaxf(red[t], red[t + s]);
        __syncthreads();
    }
    mx = red[0];
    __syncthreads();

    float sum = 0.0f;
#pragma unroll
    for (int e = 0; e < 16; ++e) {
        vals[e] = expf(vals[e] - mx);
        sum += vals[e];
    }
    red[t] = sum;
    __syncthreads();
    for (int s = 128; s > 0; s >>= 1) {
        if (t < s) red[t] += red[t + s];
        __syncthreads();
    }
    const float inv = 1.0f / red[0];
#pragma unroll
    for (int e = 0; e < 16; ++e) rp[t + e * 256] = vals[e] * inv;
}

// ---------------------------------------------------------------------------
extern "C" void kernel_launch(void* const* d_in, const int* in_sizes, int n_in,
                              void* d_out, int out_size, void* d_ws, size_t ws_size,
                              hipStream_t stream) {
    const float* gene = (const float*)d_in[0];
    const float* sp   = (const float*)d_in[1];
    const float* Wq   = (const float*)d_in[2];
    const float* bq   = (const float*)d_in[3];
    const float* Wk   = (const float*)d_in[4];
    const float* bk   = (const float*)d_in[5];
    const float* Wv   = (const float*)d_in[6];
    const float* bv   = (const float*)d_in[7];
    const float* Ws   = (const float*)d_in[8];
    const float* bs   = (const float*)d_in[9];
    const float* W1   = (const float*)d_in[10];
    const float* b1   = (const float*)d_in[11];
    const float* W2   = (const float*)d_in[12];
    const float* b2   = (const float*)d_in[13];
    const float* Wf   = (const float*)d_in[14];
    const float* bf   = (const float*)d_in[15];

    float* attn = (float*)d_out;                                  // [N, N]
    float* upd  = attn + (size_t)N_NODES * N_NODES;               // [N, OUT]

    // Workspace layout (~25 MB of fp32)
    float* ws = (float*)d_ws;
    float* Q   = ws;  ws += (size_t)N_NODES * HID;
    float* K   = ws;  ws += (size_t)N_NODES * HID;
    float* V   = ws;  ws += (size_t)N_NODES * OUT_D;
    float* si  = ws;  ws += (size_t)N_NODES * HID;
    float* h   = ws;  ws += (size_t)N_NODES * HID;
    float* e2  = ws;  ws += (size_t)N_NODES * HID;
    float* sq  = ws;  ws += N_NODES;
    float* inv = ws;  ws += N_NODES;
    float* fv  = ws;  ws += N_NODES;

    const dim3 wave(32);
    const dim3 tilesH(HID / 32, N_NODES / 32);      // [N,256] outputs
    const dim3 tilesO(OUT_D / 32, N_NODES / 32);    // [N,256] output
    const dim3 tilesL(N_NODES / 32, N_NODES / 32);  // [N,N] logits

    // Row statistics + spatial MLP front-end
    row_stats<<<N_NODES, 256, 0, stream>>>(gene, sp, sq, inv);
    spatial_in<<<(N_NODES * HID + 255) / 256, 256, 0, stream>>>(sp, Ws, bs, si);
    gemm32_wmma<1><<<tilesH, wave, 0, stream>>>(si, W1, b1, h,  HID, HID);
    gemm32_wmma<2><<<tilesH, wave, 0, stream>>>(h,  W2, b2, e2, HID, HID);
    final_proj<<<N_NODES, 256, 0, stream>>>(e2, Wf, bf, fv);

    // Q / K / V projections
    gemm32_wmma<0><<<tilesH, wave, 0, stream>>>(gene, Wq, bq, Q, GENE_D, HID);
    gemm32_wmma<0><<<tilesH, wave, 0, stream>>>(gene, Wk, bk, K, GENE_D, HID);
    gemm32_wmma<0><<<tilesO, wave, 0, stream>>>(gene, Wv, bv, V, GENE_D, OUT_D);

    // Fused logits -> softmax (in place in d_out) -> attn @ V
    logits_wmma<<<tilesL, wave, 0, stream>>>(Q, K, gene, sp, sq, inv, fv, attn);
    softmax_rows<<<N_NODES, 256, 0, stream>>>(attn);
    gemm32_wmma<0><<<tilesO, wave, 0, stream>>>(attn, V, nullptr, upd, N_NODES, OUT_D);
}